// GATedge_86474871538035
// MI455X (gfx1250) — compile-verified
//
#include <hip/hip_runtime.h>
#include <hip/hip_bf16.h>
#include <math.h>

#define B_      16
#define O_      1024
#define M_      64
#define NCHUNK  8
#define CHUNK_I (O_ / NCHUNK)   // 128
#define SLOPE_  0.2f
#define PSTRIDE 52

typedef __attribute__((ext_vector_type(2))) float v2f;
typedef __attribute__((ext_vector_type(8))) float v8f;

// ---- workspace layout (float offsets) ----
#define WS_WV_OPE   0                     // 16
#define WS_WV_MAS   16                    // 8
#define WS_WV_BUF   24                    // 8
#define WS_WV_AIN   32                    // 8
#define WS_WV_AOUT  40                    // 8
#define WS_EOPE     48                    // B*O = 16384
#define WS_EMAS     (48 + 16384)          // 1024
#define WS_EBUF     (WS_EMAS + 1024)      // 1024
#define WS_PART     (WS_EBUF + 1024)      // 2*16*8*64*52 floats
// partial record: [0]=m_in [1]=Z_in [2]=m_out [3]=Z_out
//                 [4..11]=acc_in_arc [12..27]=acc_in_ope
//                 [28..43]=acc_out_ope [44..51]=sum_arc_out

// ============ K0: fold attention vectors into weights ============
__global__ __launch_bounds__(64) void k0_wv(
    const float* __restrict__ Wope, const float* __restrict__ Wmas,
    const float* __restrict__ Wbuf, const float* __restrict__ Wain,
    const float* __restrict__ Waout,
    const float* __restrict__ a_ope, const float* __restrict__ a_mas,
    const float* __restrict__ a_arc, float* ws) {
  int t = threadIdx.x;
  const float* W; const float* a; int row; float* dst;
  if      (t < 16) { W = Wope;  a = a_ope; row = t;      dst = ws + WS_WV_OPE  + t;      }
  else if (t < 24) { W = Wmas;  a = a_mas; row = t - 16; dst = ws + WS_WV_MAS  + (t-16); }
  else if (t < 32) { W = Wbuf;  a = a_mas; row = t - 24; dst = ws + WS_WV_BUF  + (t-24); }
  else if (t < 40) { W = Wain;  a = a_arc; row = t - 32; dst = ws + WS_WV_AIN  + (t-32); }
  else if (t < 48) { W = Waout; a = a_arc; row = t - 40; dst = ws + WS_WV_AOUT + (t-40); }
  else return;
  float s = 0.f;
  #pragma unroll
  for (int n = 0; n < 32; ++n) s += W[row*32 + n] * a[n];
  *dst = s;
}

// ============ K1: per-node attention scalars ============
__global__ __launch_bounds__(256) void k1_escores(
    const float* __restrict__ feat_opes, const float* __restrict__ feat_mas,
    const float* __restrict__ feat_buf, float* ws) {
  int j = blockIdx.x * 256 + threadIdx.x;
  if (j < B_*O_) {
    const float* f = feat_opes + (size_t)j * 16;
    float s = 0.f;
    #pragma unroll
    for (int c = 0; c < 16; ++c) s += f[c] * ws[WS_WV_OPE + c];
    ws[WS_EOPE + j] = s;
  } else if (j < B_*O_ + B_*M_) {
    int r = j - B_*O_;
    const float* f = feat_mas + (size_t)r * 8;
    float s = 0.f;
    #pragma unroll
    for (int c = 0; c < 8; ++c) s += f[c] * ws[WS_WV_MAS + c];
    ws[WS_EMAS + r] = s;
  } else if (j < B_*O_ + 2*B_*M_) {
    int r = j - B_*O_ - B_*M_;
    const float* f = feat_buf + (size_t)r * 8;
    float s = 0.f;
    #pragma unroll
    for (int c = 0; c < 8; ++c) s += f[c] * ws[WS_WV_BUF + c];
    ws[WS_EBUF + r] = s;
  }
}

// ============ K2: streaming online-softmax reduction over i ============
// grid = (NCHUNK, B), block = 64 (thread = k). Reads each arc tensor once.
__global__ __launch_bounds__(64) void k2_stream(
    const float* __restrict__ adj_in, const float* __restrict__ adj_out,
    const float* __restrict__ arc_in, const float* __restrict__ arc_out,
    const float* __restrict__ feat_opes, float* ws, int branch) {
  const int c = blockIdx.x;
  const int b = blockIdx.y;
  const int k = threadIdx.x;

  const float e_node = ws[(branch ? WS_EBUF : WS_EMAS) + b*M_ + k];
  float win[8], wot[8];
  #pragma unroll
  for (int j = 0; j < 8; ++j) { win[j] = ws[WS_WV_AIN + j]; wot[j] = ws[WS_WV_AOUT + j]; }

  float m_in = -1e30f, Z_in = 0.f, m_out = -1e30f, Z_out = 0.f;
  float ain_arc[8], ain_ope[16], aout_ope[16], s_arc[8];
  #pragma unroll
  for (int j = 0; j < 8; ++j)  { ain_arc[j] = 0.f; s_arc[j] = 0.f; }
  #pragma unroll
  for (int j = 0; j < 16; ++j) { ain_ope[j] = 0.f; aout_ope[j] = 0.f; }

  const int i0 = c * CHUNK_I;
  for (int ii = 0; ii < CHUNK_I; ++ii) {
    const int i = i0 + ii;
    const float* pin  = arc_in  + (((size_t)(b*O_ + i))*M_ + k)*8;
    const float* pout = arc_out + (((size_t)(b*O_ + i))*M_ + k)*8;
    if (ii + 1 < CHUNK_I) {                       // gfx1250 global_prefetch_b8
      __builtin_prefetch(pin  + M_*8, 0, 1);
      __builtin_prefetch(pout + M_*8, 0, 1);
    }
    float4 fi0 = *(const float4*)(pin);
    float4 fi1 = *(const float4*)(pin + 4);
    float4 fo0 = *(const float4*)(pout);
    float4 fo1 = *(const float4*)(pout + 4);
    float a_in  = adj_in [i*M_ + k];
    float a_out = adj_out[i*M_ + k];
    float eope  = ws[WS_EOPE + b*O_ + i];

    const float4* fop4 = (const float4*)(feat_opes + ((size_t)b*O_ + i)*16); // uniform
    float4 p0 = fop4[0], p1 = fop4[1], p2 = fop4[2], p3 = fop4[3];
    float fo[16] = {p0.x,p0.y,p0.z,p0.w, p1.x,p1.y,p1.z,p1.w,
                    p2.x,p2.y,p2.z,p2.w, p3.x,p3.y,p3.z,p3.w};

    // unweighted sum of arc_out over ALL i (reference sums p_arc_out unmasked)
    s_arc[0]+=fo0.x; s_arc[1]+=fo0.y; s_arc[2]+=fo0.z; s_arc[3]+=fo0.w;
    s_arc[4]+=fo1.x; s_arc[5]+=fo1.y; s_arc[6]+=fo1.z; s_arc[7]+=fo1.w;

    float earc_in  = fi0.x*win[0]+fi0.y*win[1]+fi0.z*win[2]+fi0.w*win[3]
                   + fi1.x*win[4]+fi1.y*win[5]+fi1.z*win[6]+fi1.w*win[7];
    float earc_out = fo0.x*wot[0]+fo0.y*wot[1]+fo0.z*wot[2]+fo0.w*wot[3]
                   + fo1.x*wot[4]+fo1.y*wot[5]+fo1.z*wot[6]+fo1.w*wot[7];

    if (a_in == 1.0f) {
      float e = eope + earc_in + e_node;
      e = (e >= 0.f) ? e : SLOPE_ * e;
      float mn = fmaxf(m_in, e);
      float sc = __expf(m_in - mn);
      float w  = __expf(e - mn);
      m_in = mn;
      Z_in = Z_in*sc + w;
      ain_arc[0]=ain_arc[0]*sc+w*fi0.x; ain_arc[1]=ain_arc[1]*sc+w*fi0.y;
      ain_arc[2]=ain_arc[2]*sc+w*fi0.z; ain_arc[3]=ain_arc[3]*sc+w*fi0.w;
      ain_arc[4]=ain_arc[4]*sc+w*fi1.x; ain_arc[5]=ain_arc[5]*sc+w*fi1.y;
      ain_arc[6]=ain_arc[6]*sc+w*fi1.z; ain_arc[7]=ain_arc[7]*sc+w*fi1.w;
      #pragma unroll
      for (int j = 0; j < 16; ++j) ain_ope[j] = ain_ope[j]*sc + w*fo[j];
    }
    if (a_out == 1.0f) {
      float e = eope + earc_out + e_node;
      e = (e >= 0.f) ? e : SLOPE_ * e;
      float mn = fmaxf(m_out, e);
      float sc = __expf(m_out - mn);
      float w  = __expf(e - mn);
      m_out = mn;
      Z_out = Z_out*sc + w;
      #pragma unroll
      for (int j = 0; j < 16; ++j) aout_ope[j] = aout_ope[j]*sc + w*fo[j];
    }
  }

  float* p = ws + WS_PART + ((((size_t)branch*B_ + b)*NCHUNK + c)*M_ + k)*PSTRIDE;
  p[0] = m_in; p[1] = Z_in; p[2] = m_out; p[3] = Z_out;
  #pragma unroll
  for (int j = 0; j < 8; ++j)  p[4+j]  = ain_arc[j];
  #pragma unroll
  for (int j = 0; j < 16; ++j) p[12+j] = ain_ope[j];
  #pragma unroll
  for (int j = 0; j < 16; ++j) p[28+j] = aout_ope[j];
  #pragma unroll
  for (int j = 0; j < 8; ++j)  p[44+j] = s_arc[j];
}

// ============ K3: combine chunks, WMMA (rows x 40)@(40x32), sigmoid ============
// grid = 16 (branch*8 + tile), block = 256 (8 waves, each wave owns 16 rows)
__global__ __launch_bounds__(256) void k3_final(
    const float* __restrict__ feat_mas, const float* __restrict__ feat_buf,
    const float* __restrict__ Wope, const float* __restrict__ Wmas,
    const float* __restrict__ Wbuf, const float* __restrict__ Wain,
    const float* __restrict__ Waout,
    const float* ws, float* __restrict__ out) {
  __shared__ float lds_u[128][41];
  __shared__ float wmat[40][32];

  const int br   = blockIdx.x >> 3;
  const int tile = blockIdx.x & 7;
  const int t    = threadIdx.x;
  const int rowbase = tile * 128;
  const float* Wnode = br ? Wbuf : Wmas;
  const float* fnode = br ? feat_buf : feat_mas;

  // stage concatenated 40x32 weight matrix in LDS
  for (int idx = t; idx < 40*32; idx += 256) {
    int kk = idx >> 5, n = idx & 31;
    const float* src;
    if      (kk < 16) src = Wope  + kk*32;
    else if (kk < 24) src = Wain  + (kk-16)*32;
    else if (kk < 32) src = Waout + (kk-24)*32;
    else              src = Wnode + (kk-32)*32;
    wmat[kk][n] = src[n];
  }

  // phase 1: per-row chunk combine -> 40-dim reduced vector
  if (t < 128) {
    const int r = rowbase + t;          // global row = b*64+k
    float m_in = -1e30f, Z_in = 0.f, m_out = -1e30f, Z_out = 0.f;
    float ain_arc[8] = {0}, ain_ope[16] = {0}, aout_ope[16] = {0}, s_arc[8] = {0};
    for (int c = 0; c < NCHUNK; ++c) {
      const float* p = ws + WS_PART +
          ((((size_t)br*B_ + (r >> 6))*NCHUNK + c)*M_ + (r & 63))*PSTRIDE;
      float mc = p[0], Zc = p[1];
      float mn = fmaxf(m_in, mc);
      float s1 = __expf(m_in - mn), s2 = __expf(mc - mn);
      m_in = mn; Z_in = Z_in*s1 + Zc*s2;
      #pragma unroll
      for (int j = 0; j < 8;  ++j) ain_arc[j] = ain_arc[j]*s1 + p[4+j]*s2;
      #pragma unroll
      for (int j = 0; j < 16; ++j) ain_ope[j] = ain_ope[j]*s1 + p[12+j]*s2;
      float mo = p[2], Zo = p[3];
      mn = fmaxf(m_out, mo);
      s1 = __expf(m_out - mn); s2 = __expf(mo - mn);
      m_out = mn; Z_out = Z_out*s1 + Zo*s2;
      #pragma unroll
      for (int j = 0; j < 16; ++j) aout_ope[j] = aout_ope[j]*s1 + p[28+j]*s2;
      #pragma unroll
      for (int j = 0; j < 8;  ++j) s_arc[j] += p[44+j];
    }
    // self (kk) entry: e = leaky(2*e_node), always unmasked
    float en = ws[(br ? WS_EBUF : WS_EMAS) + r];
    float ek = 2.f * en; ek = (ek >= 0.f) ? ek : SLOPE_ * ek;

    float mn = fmaxf(m_in, ek);
    float s1 = __expf(m_in - mn);
    float wkk = __expf(ek - mn);
    Z_in = Z_in*s1 + wkk;
    float inv_in = 1.f / Z_in;
    float alkk_in = wkk * inv_in;
    float scale_in = s1 * inv_in;

    mn = fmaxf(m_out, ek);
    s1 = __expf(m_out - mn);
    wkk = __expf(ek - mn);
    Z_out = Z_out*s1 + wkk;
    float inv_out = 1.f / Z_out;
    float alkk_out = wkk * inv_out;
    float scale_out = s1 * inv_out;

    const float* fn = fnode + (size_t)r * 8;
    float cf = alkk_in + alkk_out;
    #pragma unroll
    for (int j = 0; j < 16; ++j) lds_u[t][j]    = ain_ope[j]*scale_in + aout_ope[j]*scale_out;
    #pragma unroll
    for (int j = 0; j < 8;  ++j) lds_u[t][16+j] = ain_arc[j]*scale_in;
    #pragma unroll
    for (int j = 0; j < 8;  ++j) lds_u[t][24+j] = s_arc[j];
    #pragma unroll
    for (int j = 0; j < 8;  ++j) lds_u[t][32+j] = cf * fn[j];
  }
  __syncthreads();

  // phase 2: WMMA f32 16x16x4, K=40 (10 steps), N=32 (2 tiles) per wave
  const int wave = t >> 5;
  const int lane = t & 31;
  const int hi   = lane >> 4;          // half-wave selects K pair (A) / K row (B)
  const int lrow = wave*16 + (lane & 15);
  const int nlo  = lane & 15;

  v8f c0 = {}; v8f c1 = {};
  #pragma unroll
  for (int k0 = 0; k0 < 40; k0 += 4) {
    v2f a;  a.x  = lds_u[lrow][k0 + 2*hi];     a.y  = lds_u[lrow][k0 + 2*hi + 1];
    v2f b0; b0.x = wmat[k0 + 2*hi][nlo];       b0.y = wmat[k0 + 2*hi + 1][nlo];
    v2f b1; b1.x = wmat[k0 + 2*hi][16 + nlo];  b1.y = wmat[k0 + 2*hi + 1][16 + nlo];
    c0 = __builtin_amdgcn_wmma_f32_16x16x4_f32(false, a, false, b0, (short)0, c0, false, false);
    c1 = __builtin_amdgcn_wmma_f32_16x16x4_f32(false, a, false, b1, (short)0, c1, false, false);
  }

  const int growb = rowbase + wave*16;
  #pragma unroll
  for (int v = 0; v < 8; ++v) {
    int m = v + 8*hi;                         // C/D layout: VGPR v holds M=v / M=v+8
    size_t base = ((size_t)br*1024 + growb + m) * 32;
    float x0 = c0[v]; x0 = 1.f / (1.f + __expf(-x0));
    float x1 = c1[v]; x1 = 1.f / (1.f + __expf(-x1));
    out[base + nlo]      = x0;
    out[base + 16 + nlo] = x1;
  }
}

extern "C" void kernel_launch(void* const* d_in, const int* in_sizes, int n_in,
                              void* d_out, int out_size, void* d_ws, size_t ws_size,
                              hipStream_t stream) {
  const float* adj0    = (const float*)d_in[0];
  const float* adj1    = (const float*)d_in[1];
  const float* adj2    = (const float*)d_in[2];
  const float* adj3    = (const float*)d_in[3];
  // d_in[4] = batch_idxes (identity arange) — unused
  const float* f_opes  = (const float*)d_in[5];
  const float* f_mas   = (const float*)d_in[6];
  const float* f_buf   = (const float*)d_in[7];
  const float* a_ma_i  = (const float*)d_in[8];
  const float* a_bf_i  = (const float*)d_in[9];
  const float* a_ma_o  = (const float*)d_in[10];
  const float* a_bf_o  = (const float*)d_in[11];
  const float* W_ope   = (const float*)d_in[12];
  const float* W_mas   = (const float*)d_in[13];
  const float* W_buf   = (const float*)d_in[14];
  const float* W_ain   = (const float*)d_in[15];
  const float* W_aout  = (const float*)d_in[16];
  const float* at_ope  = (const float*)d_in[17];
  const float* at_mas  = (const float*)d_in[18];
  const float* at_arc  = (const float*)d_in[19];
  float* ws  = (float*)d_ws;
  float* out = (float*)d_out;

  k0_wv<<<1, 64, 0, stream>>>(W_ope, W_mas, W_buf, W_ain, W_aout,
                              at_ope, at_mas, at_arc, ws);
  k1_escores<<<72, 256, 0, stream>>>(f_opes, f_mas, f_buf, ws);

  dim3 g2(NCHUNK, B_);
  k2_stream<<<g2, 64, 0, stream>>>(adj0, adj1, a_ma_i, a_ma_o, f_opes, ws, 0);
  k2_stream<<<g2, 64, 0, stream>>>(adj2, adj3, a_bf_i, a_bf_o, f_opes, ws, 1);

  k3_final<<<16, 256, 0, stream>>>(f_mas, f_buf, W_ope, W_mas, W_buf, W_ain, W_aout,
                                   ws, out);
}